// AUCMaxLoss_1142461301174
// MI455X (gfx1250) — compile-verified
//
#include <hip/hip_runtime.h>
#include <hip/hip_bf16.h>

typedef __attribute__((ext_vector_type(2))) float v2f;
typedef __attribute__((ext_vector_type(8))) float v8f;

#define NEG_BIG (-1.0e30f)
#define MARGIN 1.0f

// ---------------------------------------------------------------------------
// Kernel 1: build x[i] = pos ? (MARGIN - s_i) : -BIG ; y[j] = neg ? s_j : -BIG
// Padded region [B, Bpad) gets -BIG so padded pairs contribute exactly 0.
// ---------------------------------------------------------------------------
__global__ void auc_prep_kernel(const float* __restrict__ logits,
                                const int* __restrict__ targets,
                                float* __restrict__ x,
                                float* __restrict__ y,
                                int B, int Bpad) {
    int i = blockIdx.x * blockDim.x + threadIdx.x;
    if (i >= Bpad) return;
    if (i < B) {
        int t = targets[i];
        float s = logits[2 * i + t];
        x[i] = (t == 1) ? (MARGIN - s) : NEG_BIG;
        y[i] = (t == 0) ? s : NEG_BIG;
    } else {
        x[i] = NEG_BIG;
        y[i] = NEG_BIG;
    }
}

// ---------------------------------------------------------------------------
// Kernel 2: pairwise outer-sum tiles via V_WMMA_F32_16X16X4_F32.
// D[m][n] = x[i0+m]*1 + 1*y[j0+n] + 0  (rank-2 matmul => exact outer sum),
// then relu + accumulate. One block per i-tile (16 rows), 8 waves per block;
// each wave owns a contiguous chunk of j-tiles (scalar loop bounds via
// readfirstlane => s_cmp/s_cbranch loop, EXEC all-ones around WMMA).
// 4x unrolled: 4 independent WMMAs in flight per iteration.
// ---------------------------------------------------------------------------
__global__ void auc_pairwise_wmma_kernel(const float* __restrict__ x,
                                         const float* __restrict__ y,
                                         float* __restrict__ partials,
                                         int numTiles) {
    const int lane   = threadIdx.x & 31;
    const int lane15 = lane & 15;
    const int iTile  = blockIdx.x;

    // Scalarize the wave index so the j-loop is uniform (SALU) control flow.
    const int waveId = __builtin_amdgcn_readfirstlane((int)(threadIdx.x >> 5));

    // A fragment (16x4 f32): lanes 0-15 hold K=0 (A[0]) and K=1 (A[1]);
    // lanes 16-31 hold K=2,K=3 which stay zero (so they contribute 0*0).
    const float xv = x[iTile * 16 + lane15];
    v2f Afrag;
    Afrag[0] = (lane < 16) ? xv   : 0.0f;        // column K=0: x_i
    Afrag[1] = (lane < 16) ? 1.0f : 0.0f;        // column K=1: ones

    // Constant B row K=0 (ones in lanes 0-15), hoisted out of the loop.
    const float bRow0 = (lane < 16) ? 1.0f : 0.0f;
    const float yMask = bRow0;                   // same mask for y row K=1

    // Contiguous chunk of j-tiles per wave.
    const int tilesPerWave = (numTiles + 7) / 8;
    const int jBegin = waveId * tilesPerWave;
    int jEnd = jBegin + tilesPerWave;
    if (jEnd > numTiles) jEnd = numTiles;

    float acc0 = 0.0f, acc1 = 0.0f, acc2 = 0.0f, acc3 = 0.0f;
    int j = jBegin;
    for (; j + 4 <= jEnd; j += 4) {
        const float y0 = y[(j + 0) * 16 + lane15] * yMask;
        const float y1 = y[(j + 1) * 16 + lane15] * yMask;
        const float y2 = y[(j + 2) * 16 + lane15] * yMask;
        const float y3 = y[(j + 3) * 16 + lane15] * yMask;
        v2f B0; B0[0] = bRow0; B0[1] = y0;
        v2f B1; B1[0] = bRow0; B1[1] = y1;
        v2f B2; B2[0] = bRow0; B2[1] = y2;
        v2f B3; B3[0] = bRow0; B3[1] = y3;
        v8f Cz = {};
        v8f D0 = __builtin_amdgcn_wmma_f32_16x16x4_f32(false, Afrag, false, B0, (short)0, Cz, false, false);
        v8f D1 = __builtin_amdgcn_wmma_f32_16x16x4_f32(false, Afrag, false, B1, (short)0, Cz, false, false);
        v8f D2 = __builtin_amdgcn_wmma_f32_16x16x4_f32(false, Afrag, false, B2, (short)0, Cz, false, false);
        v8f D3 = __builtin_amdgcn_wmma_f32_16x16x4_f32(false, Afrag, false, B3, (short)0, Cz, false, false);
#pragma unroll
        for (int k = 0; k < 8; ++k) {
            acc0 += fmaxf(D0[k], 0.0f);
            acc1 += fmaxf(D1[k], 0.0f);
            acc2 += fmaxf(D2[k], 0.0f);
            acc3 += fmaxf(D3[k], 0.0f);
        }
    }
    for (; j < jEnd; ++j) {                      // remainder (scalar-bound loop)
        const float yv = y[j * 16 + lane15] * yMask;
        v2f Bf; Bf[0] = bRow0; Bf[1] = yv;
        v8f Cz = {};
        v8f D = __builtin_amdgcn_wmma_f32_16x16x4_f32(false, Afrag, false, Bf, (short)0, Cz, false, false);
#pragma unroll
        for (int k = 0; k < 8; ++k) acc0 += fmaxf(D[k], 0.0f);
    }

    float acc = (acc0 + acc1) + (acc2 + acc3);

    // wave32 reduction (fixed order -> deterministic)
#pragma unroll
    for (int off = 16; off > 0; off >>= 1) {
        acc += __shfl_xor(acc, off, 32);
    }

    __shared__ float wsum[8];
    if (lane == 0) wsum[waveId] = acc;
    __syncthreads();
    if (threadIdx.x == 0) {
        float s = 0.0f;
#pragma unroll
        for (int w = 0; w < 8; ++w) s += wsum[w];
        partials[iTile] = s;
    }
}

// ---------------------------------------------------------------------------
// Kernel 3: deterministic final reduction + normalization.
// ---------------------------------------------------------------------------
__global__ void auc_finalize_kernel(const float* __restrict__ partials,
                                    int numTiles,
                                    const int* __restrict__ targets,
                                    int B,
                                    float* __restrict__ out) {
    __shared__ float ssum[256];
    __shared__ float scnt[256];
    float s = 0.0f, c = 0.0f;
    for (int i = threadIdx.x; i < numTiles; i += 256) s += partials[i];
    for (int i = threadIdx.x; i < B; i += 256) c += (float)targets[i];
    ssum[threadIdx.x] = s;
    scnt[threadIdx.x] = c;
    __syncthreads();
    for (int st = 128; st > 0; st >>= 1) {
        if (threadIdx.x < st) {
            ssum[threadIdx.x] += ssum[threadIdx.x + st];
            scnt[threadIdx.x] += scnt[threadIdx.x + st];
        }
        __syncthreads();
    }
    if (threadIdx.x == 0) {
        float npos = scnt[0];
        float nneg = (float)B - npos;
        out[0] = ssum[0] / (npos * nneg + 1e-8f);
    }
}

// ---------------------------------------------------------------------------
extern "C" void kernel_launch(void* const* d_in, const int* in_sizes, int n_in,
                              void* d_out, int out_size, void* d_ws, size_t ws_size,
                              hipStream_t stream) {
    const float* logits  = (const float*)d_in[0];
    const int*   targets = (const int*)d_in[1];
    const int B        = in_sizes[1];           // 16384
    const int numTiles = (B + 15) / 16;         // 1024
    const int Bpad     = numTiles * 16;

    float* x        = (float*)d_ws;             // [Bpad]
    float* y        = x + Bpad;                 // [Bpad]
    float* partials = y + Bpad;                 // [numTiles]

    auc_prep_kernel<<<(Bpad + 255) / 256, 256, 0, stream>>>(
        logits, targets, x, y, B, Bpad);

    auc_pairwise_wmma_kernel<<<numTiles, 256, 0, stream>>>(
        x, y, partials, numTiles);

    auc_finalize_kernel<<<1, 256, 0, stream>>>(
        partials, numTiles, targets, B, (float*)d_out);
}